// GATLayer_27376121545165
// MI455X (gfx1250) — compile-verified
//
#include <hip/hip_runtime.h>

typedef float v2f __attribute__((ext_vector_type(2)));
typedef float v8f __attribute__((ext_vector_type(8)));

#define IN_FEATS  256
#define OUT_FEATS 256
#define HEADS     4
#define D_OUT     64
#define NEG_SLOPE 0.2f
#define ROWS_BLK  32    // rows per block (2 x 16-row WMMA tiles)
#define LDS_PAD   260   // 256 + 4: stride 260 % 64 == 4 -> conflict-free column reads

// ---------------------------------------------------------------------------
// Kernel 1: P = F @ W via V_WMMA_F32_16X16X4_F32 (fp32 matrix path).
// Block = 256 threads = 8 waves. Block owns a 32-row panel; wave w owns output
// columns [w*32, w*32+32) for BOTH row tiles -> 4 accumulator tiles per wave,
// so each B fragment load feeds 2 WMMAs. A-panel staged in LDS (33 KB).
// W L2 traffic: (N/32) blocks * 256 KB = 800 MB (vs 1.6 GB with 16-row panels).
// ---------------------------------------------------------------------------
__global__ __launch_bounds__(256) void gat_gemm_wmma(const float* __restrict__ F,
                                                     const float* __restrict__ W,
                                                     float* __restrict__ P, int N)
{
    __shared__ __align__(16) float As[ROWS_BLK][LDS_PAD];
    const int tid  = threadIdx.x;
    const int row0 = blockIdx.x * ROWS_BLK;

    // Cooperative load: 32 rows x 256 K values (float4, coalesced along K).
    #pragma unroll
    for (int i = tid; i < ROWS_BLK * (IN_FEATS / 4); i += 256) {
        const int r  = i >> 6;              // / 64 float4s per row
        const int c4 = (i & 63) << 2;
        int rg = row0 + r; if (rg >= N) rg = N - 1;   // clamp (tail block)
        const float4 v = *(const float4*)(F + (size_t)rg * IN_FEATS + c4);
        *(float4*)(&As[r][c4]) = v;
    }
    __syncthreads();

    const int wave    = tid >> 5;
    const int lane    = tid & 31;
    const int colbase = wave * 32;
    const int mrow    = lane & 15;
    const int khalf   = (lane >> 4) << 1;   // lanes 0-15 -> K+0/1, lanes 16-31 -> K+2/3

    v8f c00 = {}; v8f c01 = {};   // rows 0-15,  cols +0 / +16
    v8f c10 = {}; v8f c11 = {};   // rows 16-31, cols +0 / +16
    #pragma unroll 2
    for (int k = 0; k < IN_FEATS; k += 4) {
        const int kk = k + khalf;
        v2f a0; a0.x = As[mrow][kk];      a0.y = As[mrow][kk + 1];
        v2f a1; a1.x = As[mrow + 16][kk]; a1.y = As[mrow + 16][kk + 1];
        const float* wp = W + (size_t)kk * OUT_FEATS + colbase + mrow;
        v2f b0; b0.x = wp[0];  b0.y = wp[OUT_FEATS];
        v2f b1; b1.x = wp[16]; b1.y = wp[OUT_FEATS + 16];
        c00 = __builtin_amdgcn_wmma_f32_16x16x4_f32(false, a0, false, b0, (short)0, c00, false, false);
        c01 = __builtin_amdgcn_wmma_f32_16x16x4_f32(false, a0, false, b1, (short)0, c01, false, false);
        c10 = __builtin_amdgcn_wmma_f32_16x16x4_f32(false, a1, false, b0, (short)0, c10, false, false);
        c11 = __builtin_amdgcn_wmma_f32_16x16x4_f32(false, a1, false, b1, (short)0, c11, false, false);
    }

    // C/D layout: VGPR r -> (M=r, N=lane) lanes 0-15; (M=r+8, N=lane-16) lanes 16-31.
    const int radd = (lane >> 4) << 3;
    const int col  = colbase + mrow;
    if (row0 + ROWS_BLK <= N) {
        // Uniform fast path: no per-row exec-mask churn.
        #pragma unroll
        for (int r = 0; r < 8; ++r) {
            float* p0 = P + (size_t)(row0 + r + radd) * OUT_FEATS + col;
            p0[0]  = c00[r];
            p0[16] = c01[r];
            float* p1 = p0 + 16 * OUT_FEATS;
            p1[0]  = c10[r];
            p1[16] = c11[r];
        }
    } else {
        #pragma unroll
        for (int r = 0; r < 8; ++r) {
            const int rowg = row0 + r + radd;
            if (rowg < N) {
                P[(size_t)rowg * OUT_FEATS + col]      = c00[r];
                P[(size_t)rowg * OUT_FEATS + col + 16] = c01[r];
            }
            if (rowg + 16 < N) {
                P[(size_t)(rowg + 16) * OUT_FEATS + col]      = c10[r];
                P[(size_t)(rowg + 16) * OUT_FEATS + col + 16] = c11[r];
            }
        }
    }
}

// ---------------------------------------------------------------------------
// Kernel 2: per-node attention logits  out[n,h] = sum_d P[n,h,d] * attn[h,d]
// ---------------------------------------------------------------------------
__global__ void gat_logits(const float* __restrict__ P, const float* __restrict__ attn,
                           float* __restrict__ out, int N)
{
    const int t = blockIdx.x * blockDim.x + threadIdx.x;
    if (t >= N * HEADS) return;
    const int n = t >> 2, h = t & 3;
    const float* p = P + (size_t)n * OUT_FEATS + h * D_OUT;
    const float* a = attn + h * D_OUT;
    float s = 0.f;
    #pragma unroll
    for (int d = 0; d < D_OUT; d += 4) {
        const float4 pv = *(const float4*)(p + d);
        const float4 av = *(const float4*)(a + d);
        s += pv.x * av.x + pv.y * av.y + pv.z * av.z + pv.w * av.w;
    }
    out[t] = s;
}

// ---------------------------------------------------------------------------
// Kernel 3: per-edge softmax numerator s = exp(leaky_relu(el[src]+er[dst]))
// and denominator segment-sum via hardware f32 atomics (denom is 1.6 MB -> L2).
// ---------------------------------------------------------------------------
__global__ void gat_edge_num(const int* __restrict__ es, const int* __restrict__ ed,
                             const float* __restrict__ el, const float* __restrict__ er,
                             float* __restrict__ s, float* __restrict__ denom, int E)
{
    const int e = blockIdx.x * blockDim.x + threadIdx.x;
    if (e >= E) return;
    const int u = es[e], v = ed[e];
    #pragma unroll
    for (int h = 0; h < HEADS; ++h) {
        float x = el[u * HEADS + h] + er[v * HEADS + h];
        x = (x >= 0.f) ? x : NEG_SLOPE * x;
        const float sx = expf(x);
        s[(size_t)e * HEADS + h] = sx;
        unsafeAtomicAdd(&denom[v * HEADS + h], sx);
    }
}

// ---------------------------------------------------------------------------
// Kernel 4: wave-per-edge aggregation. Psrc (102 MB) and out (102 MB) both fit
// in the 192 MB L2, so the 1 KB gather + 1 KB atomic scatter per edge is
// L2-resident. Lane-contiguous indices -> 128 B coalesced accesses.
// ---------------------------------------------------------------------------
__global__ __launch_bounds__(256) void gat_aggregate(const int* __restrict__ es,
                                                     const int* __restrict__ ed,
                                                     const float* __restrict__ s,
                                                     const float* __restrict__ denom,
                                                     const float* __restrict__ Psrc,
                                                     float* __restrict__ out, int E)
{
    const int wid  = (int)((blockIdx.x * 256u + threadIdx.x) >> 5);
    const int lane = threadIdx.x & 31;
    if (wid >= E) return;
    const int u = es[wid], v = ed[wid];
    float a[HEADS];
    #pragma unroll
    for (int h = 0; h < HEADS; ++h)
        a[h] = s[(size_t)wid * HEADS + h] / denom[v * HEADS + h];
    const float* ps = Psrc + (size_t)u * OUT_FEATS;
    float*       po = out  + (size_t)v * OUT_FEATS;
    #pragma unroll
    for (int i = 0; i < 8; ++i) {
        const int idx = lane + i * 32;
        unsafeAtomicAdd(&po[idx], ps[idx] * a[idx >> 6]);
    }
}

// ---------------------------------------------------------------------------
extern "C" void kernel_launch(void* const* d_in, const int* in_sizes, int n_in,
                              void* d_out, int out_size, void* d_ws, size_t ws_size,
                              hipStream_t stream)
{
    const float* feat_src = (const float*)d_in[0];
    const float* feat_dst = (const float*)d_in[1];
    const int*   edge_src = (const int*)  d_in[2];
    const int*   edge_dst = (const int*)  d_in[3];
    const float* W        = (const float*)d_in[4];
    const float* attn_l   = (const float*)d_in[5];
    const float* attn_r   = (const float*)d_in[6];

    const int Ns = in_sizes[0] / IN_FEATS;
    const int Nd = in_sizes[1] / IN_FEATS;
    const int E  = in_sizes[2];

    // Workspace layout (floats)
    float* Psrc  = (float*)d_ws;
    float* Pdst  = Psrc + (size_t)Ns * OUT_FEATS;
    float* el    = Pdst + (size_t)Nd * OUT_FEATS;
    float* er    = el   + (size_t)Ns * HEADS;
    float* sbuf  = er   + (size_t)Nd * HEADS;
    float* denom = sbuf + (size_t)E  * HEADS;

    hipMemsetAsync(denom, 0, (size_t)Nd * HEADS * sizeof(float), stream);
    hipMemsetAsync(d_out, 0, (size_t)out_size * sizeof(float), stream);

    const dim3 blk(256);
    gat_gemm_wmma<<<(Ns + ROWS_BLK - 1) / ROWS_BLK, blk, 0, stream>>>(feat_src, W, Psrc, Ns);
    gat_gemm_wmma<<<(Nd + ROWS_BLK - 1) / ROWS_BLK, blk, 0, stream>>>(feat_dst, W, Pdst, Nd);
    gat_logits<<<(Ns * HEADS + 255) / 256, blk, 0, stream>>>(Psrc, attn_l, el, Ns);
    gat_logits<<<(Nd * HEADS + 255) / 256, blk, 0, stream>>>(Pdst, attn_r, er, Nd);
    gat_edge_num<<<(E + 255) / 256, blk, 0, stream>>>(edge_src, edge_dst, el, er, sbuf, denom, E);
    gat_aggregate<<<(E + 7) / 8, blk, 0, stream>>>(edge_src, edge_dst, sbuf, denom, Psrc,
                                                   (float*)d_out, E);
}